// MaskGenerator_72035191489122
// MI455X (gfx1250) — compile-verified
//
#include <hip/hip_runtime.h>
#include <hip/hip_bf16.h>
#include <math.h>

// ---------------------------------------------------------------------------
// CDNA5 (gfx1250) fp32 WMMA types
// ---------------------------------------------------------------------------
typedef __attribute__((ext_vector_type(2))) float v2f;
typedef __attribute__((ext_vector_type(8))) float v8f;

// ---------------------------------------------------------------------------
// Degree / normalization kernels
// ---------------------------------------------------------------------------
__global__ void deg_init_kernel(float* __restrict__ deg, int n) {
    int i = blockIdx.x * blockDim.x + threadIdx.x;
    if (i < n) deg[i] = 1.0f;  // self-loop contribution
}

__global__ void deg_edge_kernel(const int* __restrict__ dst, float* __restrict__ deg, int E) {
    int e = blockIdx.x * blockDim.x + threadIdx.x;
    if (e < E) atomicAdd(&deg[dst[e]], 1.0f);
}

__global__ void dinv_kernel(float* __restrict__ deg, int n) {
    int i = blockIdx.x * blockDim.x + threadIdx.x;
    if (i < n) {
        float d = deg[i];
        deg[i] = (d > 0.0f) ? rsqrtf(d) : 0.0f;
    }
}

// ---------------------------------------------------------------------------
// fp32 WMMA GEMM: Out[M,N] = epilogue(X[M,K] @ W[K,N] + bias)
// One wave computes a 32(M) x 64(N) tile: 2 A fragments x 4 B fragments,
// 8 v8f accumulators, K-step = 4 (v_wmma_f32_16x16x4_f32).
// EPI: 0 = none, 1 = bias+relu, 2 = bias+sigmoid (+ nontemporal store)
// A layout (32-bit 16x4): lanes 0-15 -> M=lane, v0=K0 v1=K1;
//                         lanes 16-31 -> M=lane-16, v0=K2 v1=K3.
// B layout (32-bit 4x16): lanes 0-15 -> N=lane, v0=K0 v1=K1;
//                         lanes 16-31 -> N=lane-16, v0=K2 v1=K3.
// C/D layout: vgpr i -> M = i (+8 for lanes 16-31), N = lane&15 (+n-tile).
// ---------------------------------------------------------------------------
template <int EPI>
__global__ __launch_bounds__(256) void gemm_wmma_kernel(
    const float* __restrict__ X, const float* __restrict__ W,
    const float* __restrict__ bias, float* __restrict__ Out,
    int M, int K, int Nout, int mTiles32, int nTiles64) {
    int wave = (blockIdx.x * blockDim.x + threadIdx.x) >> 5;  // wave-uniform
    int lane = threadIdx.x & 31;
    if (wave >= mTiles32 * nTiles64) return;  // uniform exit: EXEC stays all-1s

    int tm = wave % mTiles32;
    int tn = wave / mTiles32;
    int m0 = tm * 32;
    int n0 = tn * 64;

    int row  = lane & 15;
    int half = lane >> 4;  // 0: K0/K1, 1: K2/K3

    int ar0 = m0 + row;      if (ar0 >= M) ar0 = M - 1;  // clamp loads
    int ar1 = m0 + 16 + row; if (ar1 >= M) ar1 = M - 1;
    const float* aPtr0 = X + (size_t)ar0 * K + 2 * half;
    const float* aPtr1 = X + (size_t)ar1 * K + 2 * half;

    v8f acc[2][4] = {};

    for (int k = 0; k < K; k += 4) {
        v2f a0, a1;
        a0.x = aPtr0[k];
        a0.y = aPtr0[k + 1];
        a1.x = aPtr1[k];
        a1.y = aPtr1[k + 1];

        const float* wp0 = W + (size_t)(k + 2 * half) * Nout + n0 + row;
        const float* wp1 = wp0 + Nout;
        v2f b0, b1, b2, b3;
        b0.x = wp0[0];  b0.y = wp1[0];
        b1.x = wp0[16]; b1.y = wp1[16];
        b2.x = wp0[32]; b2.y = wp1[32];
        b3.x = wp0[48]; b3.y = wp1[48];

        acc[0][0] = __builtin_amdgcn_wmma_f32_16x16x4_f32(false, a0, false, b0, (short)0, acc[0][0], false, false);
        acc[0][1] = __builtin_amdgcn_wmma_f32_16x16x4_f32(false, a0, false, b1, (short)0, acc[0][1], false, false);
        acc[0][2] = __builtin_amdgcn_wmma_f32_16x16x4_f32(false, a0, false, b2, (short)0, acc[0][2], false, false);
        acc[0][3] = __builtin_amdgcn_wmma_f32_16x16x4_f32(false, a0, false, b3, (short)0, acc[0][3], false, false);
        acc[1][0] = __builtin_amdgcn_wmma_f32_16x16x4_f32(false, a1, false, b0, (short)0, acc[1][0], false, false);
        acc[1][1] = __builtin_amdgcn_wmma_f32_16x16x4_f32(false, a1, false, b1, (short)0, acc[1][1], false, false);
        acc[1][2] = __builtin_amdgcn_wmma_f32_16x16x4_f32(false, a1, false, b2, (short)0, acc[1][2], false, false);
        acc[1][3] = __builtin_amdgcn_wmma_f32_16x16x4_f32(false, a1, false, b3, (short)0, acc[1][3], false, false);
    }

#pragma unroll
    for (int mm = 0; mm < 2; ++mm) {
#pragma unroll
        for (int t = 0; t < 4; ++t) {
            int c = n0 + t * 16 + row;
#pragma unroll
            for (int i = 0; i < 8; ++i) {
                int r = m0 + mm * 16 + half * 8 + i;
                if (r < M) {
                    float v = acc[mm][t][i];
                    if (EPI != 0) v += bias[c];
                    if (EPI == 1) v = v > 0.0f ? v : 0.0f;
                    if (EPI == 2) {
                        v = 1.0f / (1.0f + expf(-v));
                        // write-once streaming output: keep it out of L2's way
                        __builtin_nontemporal_store(v, &Out[(size_t)r * Nout + c]);
                    } else {
                        Out[(size_t)r * Nout + c] = v;
                    }
                }
            }
        }
    }
}

// ---------------------------------------------------------------------------
// out[i,f] = h[i,f] * dinv[i]^2 + b_gcn[f]   (self-loop term + bias)
// ---------------------------------------------------------------------------
__global__ void out_init_kernel(const float* __restrict__ h, const float* __restrict__ dinv,
                                const float* __restrict__ b, float* __restrict__ out,
                                int Nn, int H) {
    size_t tid = (size_t)blockIdx.x * blockDim.x + threadIdx.x;
    size_t total = (size_t)Nn * H;
    if (tid >= total) return;
    int i = (int)(tid / H);
    int f = (int)(tid % H);
    float di = dinv[i];
    out[tid] = h[tid] * di * di + b[f];
}

// ---------------------------------------------------------------------------
// Edge scatter: out[dst] += h[src] * dinv[src]*dinv[dst]
// One wave per edge (H4==32): lanes load consecutive float4s of h[src]
// (coalesced 512B) and issue 4 fp32 atomics each to consecutive addresses.
// ---------------------------------------------------------------------------
__global__ void scatter_kernel(const float* __restrict__ h, const int* __restrict__ src,
                               const int* __restrict__ dst, const float* __restrict__ dinv,
                               float* __restrict__ out, int E, int H4) {
    int tid = blockIdx.x * blockDim.x + threadIdx.x;
    int e = tid / H4;
    if (e >= E) return;
    int q = (tid - e * H4) * 4;
    int s = src[e];
    int d = dst[e];
    float norm = dinv[s] * dinv[d];
    const float4 hv = *(const float4*)(h + (size_t)s * (H4 * 4) + q);
    float* op = out + (size_t)d * (H4 * 4) + q;
    atomicAdd(op + 0, hv.x * norm);
    atomicAdd(op + 1, hv.y * norm);
    atomicAdd(op + 2, hv.z * norm);
    atomicAdd(op + 3, hv.w * norm);
}

// ---------------------------------------------------------------------------
// Edge-score head: res[r] = sigmoid( [out[a]; out[b]] . W_adj + b_adj )
// One wave32 per row; lane covers 4 features per half; shfl_xor reduction.
// ---------------------------------------------------------------------------
__global__ __launch_bounds__(256) void adj_kernel(
    const float* __restrict__ out, const int* __restrict__ khop,
    const int* __restrict__ neg, const float* __restrict__ Wadj,
    const float* __restrict__ badj, float* __restrict__ res,
    int EK, int EN, int H) {
    int gw = (blockIdx.x * blockDim.x + threadIdx.x) >> 5;
    int lane = threadIdx.x & 31;
    if (gw >= EK + EN) return;

    int a, b;
    if (gw < EK) {
        a = khop[gw];
        b = khop[EK + gw];
    } else {
        int r = gw - EK;
        a = neg[2 * r];
        b = neg[2 * r + 1];
    }

    const float4 va = *((const float4*)(out + (size_t)a * H) + lane);
    const float4 vb = *((const float4*)(out + (size_t)b * H) + lane);
    const float4 ca = *((const float4*)(Wadj) + lane);
    const float4 cb = *((const float4*)(Wadj + H) + lane);

    float sum = va.x * ca.x + va.y * ca.y + va.z * ca.z + va.w * ca.w +
                vb.x * cb.x + vb.y * cb.y + vb.z * cb.z + vb.w * cb.w;
    for (int off = 16; off > 0; off >>= 1) sum += __shfl_xor(sum, off, 32);

    if (lane == 0) {
        float v = 1.0f / (1.0f + expf(-(sum + badj[0])));
        __builtin_nontemporal_store(v, &res[gw]);
    }
}

// ---------------------------------------------------------------------------
// Host launch
// ---------------------------------------------------------------------------
static inline unsigned cdiv(size_t a, size_t b) { return (unsigned)((a + b - 1) / b); }

extern "C" void kernel_launch(void* const* d_in, const int* in_sizes, int n_in,
                              void* d_out, int out_size, void* d_ws, size_t ws_size,
                              hipStream_t stream) {
    const float* x     = (const float*)d_in[0];
    const int*   ei    = (const int*)d_in[1];
    const int*   khop  = (const int*)d_in[2];
    const int*   neg   = (const int*)d_in[3];
    const float* Wgcn  = (const float*)d_in[4];
    const float* bgcn  = (const float*)d_in[5];
    const float* W1    = (const float*)d_in[6];
    const float* b1    = (const float*)d_in[7];
    const float* W2    = (const float*)d_in[8];
    const float* b2    = (const float*)d_in[9];
    const float* Wadj  = (const float*)d_in[10];
    const float* badj  = (const float*)d_in[11];

    const int H  = in_sizes[5];           // 128
    const int F  = in_sizes[4] / H;       // 256
    const int Nn = in_sizes[0] / F;       // 50000
    const int E  = in_sizes[1] / 2;       // 800000
    const int EK = in_sizes[2] / 2;       // 200000
    const int EN = in_sizes[3] / 2;       // 200000

    float* ws   = (float*)d_ws;
    float* dinv = ws;                       // [Nn]
    float* h    = ws + Nn;                  // [Nn*H]  (reused as t after scatter)
    float* outb = h + (size_t)Nn * H;       // [Nn*H]
    float* t    = h;                        // reuse: h dead after out is built

    float* fo  = (float*)d_out;             // feature_out [Nn*F]
    float* ao  = fo + (size_t)Nn * F;       // adj_out [EK+EN]

    const int* esrc = ei;
    const int* edst = ei + E;

    // 1) degree -> dinv
    deg_init_kernel<<<cdiv(Nn, 256), 256, 0, stream>>>(dinv, Nn);
    deg_edge_kernel<<<cdiv(E, 256), 256, 0, stream>>>(edst, dinv, E);
    dinv_kernel<<<cdiv(Nn, 256), 256, 0, stream>>>(dinv, Nn);

    // 2) h = x @ W_gcn   [Nn,F] @ [F,H]
    {
        int mT = (Nn + 31) / 32, nT = H / 64;
        gemm_wmma_kernel<0><<<cdiv((size_t)mT * nT, 8), 256, 0, stream>>>(
            x, Wgcn, nullptr, h, Nn, F, H, mT, nT);
    }

    // 3) out = self-loop term + bias, then scatter edges
    out_init_kernel<<<cdiv((size_t)Nn * H, 256), 256, 0, stream>>>(h, dinv, bgcn, outb, Nn, H);
    scatter_kernel<<<cdiv((size_t)E * (H / 4), 256), 256, 0, stream>>>(
        h, esrc, edst, dinv, outb, E, H / 4);

    // 4) t = relu(out @ W1 + b1)   [Nn,H] @ [H,H]
    {
        int mT = (Nn + 31) / 32, nT = H / 64;
        gemm_wmma_kernel<1><<<cdiv((size_t)mT * nT, 8), 256, 0, stream>>>(
            outb, W1, b1, t, Nn, H, H, mT, nT);
    }

    // 5) feature_out = sigmoid(t @ W2 + b2)   [Nn,H] @ [H,F] -> d_out (NT stores)
    {
        int mT = (Nn + 31) / 32, nT = F / 64;
        gemm_wmma_kernel<2><<<cdiv((size_t)mT * nT, 8), 256, 0, stream>>>(
            t, W2, b2, fo, Nn, H, F, mT, nT);
    }

    // 6) adj_out = sigmoid([out[a];out[b]] . W_adj + b_adj)
    adj_kernel<<<cdiv((size_t)(EK + EN), 8), 256, 0, stream>>>(
        outb, khop, neg, Wadj, badj, ao, EK, EN, H);
}